// FSGC_54494545051898
// MI455X (gfx1250) — compile-verified
//
#include <hip/hip_runtime.h>

// Problem constants (match reference)
#define N_  32
#define C_  64
#define T_  256
#define V_  25
#define R_  8
#define O_  64
#define K_  5
#define NU_ 17

typedef float v2f __attribute__((ext_vector_type(2)));
typedef float v8f __attribute__((ext_vector_type(8)));

// LDS byte offset of a generic pointer known to live in LDS
__device__ __forceinline__ unsigned lds_off(const void* p) {
  return (unsigned)(uintptr_t)(const __attribute__((address_space(3))) void*)p;
}

// ---------------------------------------------------------------------------
// Kernel A: temporal mean pool  pooled[n][c][v] = mean_t x[n][c][t][v]
// ---------------------------------------------------------------------------
__global__ __launch_bounds__(256) void pool_kernel(const float* __restrict__ x,
                                                   float* __restrict__ pooled) {
  const int nc  = blockIdx.x;          // n*C + c
  const int tid = threadIdx.x;
  const int lane = tid & 31, grp = tid >> 5;   // 8 groups of 32
  __shared__ float part[8 * 32];
  float acc = 0.f;
  if (lane < V_) {
    const float* p = x + (size_t)nc * (T_ * V_);
    for (int t = grp; t < T_; t += 8) acc += p[t * V_ + lane];
  }
  part[grp * 32 + lane] = acc;
  __syncthreads();
  if (tid < V_) {
    float s = 0.f;
    #pragma unroll
    for (int g = 0; g < 8; ++g) s += part[g * 32 + tid];
    pooled[nc * V_ + tid] = s * (1.0f / T_);
  }
}

// ---------------------------------------------------------------------------
// Kernel B: topology branch -> graphs (K,N,O,25,25)
// Branch k uses upper rows (i<17) from branch (k+1)%5 (rotation), lower rows
// (i>=17) from branch k. Columns always use W2 of the corresponding branch.
// ---------------------------------------------------------------------------
__global__ __launch_bounds__(256) void graphs_kernel(
    const float* __restrict__ pooled,
    const float* __restrict__ Ag,
    const float* __restrict__ W11, const float* __restrict__ b11,
    const float* __restrict__ W12, const float* __restrict__ b12,
    const float* __restrict__ W2,  const float* __restrict__ b2,
    const float* __restrict__ W4,  const float* __restrict__ b4,
    float* __restrict__ graphs) {
  const int n = blockIdx.x;
  const int k = blockIdx.y;
  const int k2 = (k + 1) % K_;
  const int tid = threadIdx.x;

  __shared__ float ps[C_ * V_];
  __shared__ float rowU[R_ * V_], colU[R_ * V_], rowL[R_ * 8], colL[R_ * V_];
  __shared__ float fullm[R_ * V_ * V_];
  __shared__ float w4s[O_ * R_], b4s[O_];

  for (int i = tid; i < C_ * V_; i += 256) ps[i] = pooled[n * C_ * V_ + i];
  for (int i = tid; i < O_ * R_; i += 256) w4s[i] = W4[k * O_ * R_ + i];
  if (tid < O_) b4s[tid] = b4[k * O_ + tid];
  __syncthreads();

  if (tid < R_ * V_) {
    const int r = tid / V_, j = tid % V_;
    float aU = b11[k2 * R_ + r];
    float cU = b2[k2 * R_ + r];
    float cL = b2[k * R_ + r];
    const float* w11 = W11 + (k2 * R_ + r) * C_;
    const float* w2u = W2 + (k2 * R_ + r) * C_;
    const float* w2l = W2 + (k * R_ + r) * C_;
    for (int c = 0; c < C_; ++c) {
      const float xv = ps[c * V_ + j];
      aU += w11[c] * xv;
      cU += w2u[c] * xv;
      cL += w2l[c] * xv;
    }
    rowU[tid] = aU; colU[tid] = cU; colL[tid] = cL;
  }
  if (tid < R_ * 8) {
    const int r = tid / 8, jj = tid % 8;
    float aL = b12[k * R_ + r];
    const float* w12 = W12 + (k * R_ + r) * C_;
    for (int c = 0; c < C_; ++c) aL += w12[c] * ps[c * V_ + NU_ + jj];
    rowL[tid] = aL;
  }
  __syncthreads();

  for (int idx = tid; idx < R_ * V_ * V_; idx += 256) {
    const int r = idx / (V_ * V_), rem = idx % (V_ * V_);
    const int i = rem / V_, j = rem % V_;
    const float d = (i < NU_) ? (rowU[r * V_ + i] - colU[r * V_ + j])
                              : (rowL[r * 8 + (i - NU_)] - colL[r * V_ + j]);
    fullm[idx] = tanhf(d);
  }
  __syncthreads();

  float* gout = graphs + ((size_t)(k * N_ + n)) * O_ * V_ * V_;
  for (int idx = tid; idx < O_ * V_ * V_; idx += 256) {
    const int o = idx / (V_ * V_), pos = idx % (V_ * V_);
    float acc = b4s[o];
    #pragma unroll
    for (int r = 0; r < R_; ++r) acc += w4s[o * R_ + r] * fullm[r * V_ * V_ + pos];
    gout[idx] = acc;
  }
}

// ---------------------------------------------------------------------------
// Kernel C: fused  x2 = W3 @ x (+b3);  out += x2 @ (graphs + A)^T  over k
// Block = (n, 16-t tile), 256 threads = 8 waves. All WMMA f32 16x16x4.
//
//  LDS (floats):
//    Xs2 [400][68]   x[n,:,t-tile,:] TRANSPOSED to [tv][c]; pad 68 kills
//                    bank conflicts; c-pairs adjacent for ds_load_b64 B-frags.
//    x2s [16][16][28] per (k,o-chunk) x2 tile [o][t][v], v padded to 28
//    gs2 [16][25][28] (graphs+A) as [o][u][v], v padded to 28 (+256 guard)
//    w3s [16][68]     W3 chunk, padded stride
//  Xs2 is staged with CDNA5 async global->LDS loads (ASYNCcnt).
// ---------------------------------------------------------------------------
__global__ __launch_bounds__(256) void fused_gemm_kernel(
    const float* __restrict__ x,       // (N,C,T,V)
    const float* __restrict__ graphs,  // (K,N,O,25,25)
    const float* __restrict__ Ag,      // (K,25,25)
    const float* __restrict__ W3,      // (K,O,C)
    const float* __restrict__ b3,      // (K,O)
    float* __restrict__ out) {         // (N,O,T,V)
  const int TT = 16;
  const int n = blockIdx.y;
  const int t0 = blockIdx.x * TT;

  extern __shared__ float smem[];
  float* Xs2 = smem;                          // 400*68 = 27200
  float* x2s = Xs2 + 400 * 68;                // 7168
  float* gs2 = x2s + 16 * TT * 28;            // 11200 + 256 guard
  float* w3s = gs2 + 16 * V_ * 28 + 256;      // 1088

  const int tid  = threadIdx.x;
  const int lane = tid & 31;
  const int wv   = tid >> 5;
  const int lh   = lane & 15;    // M index within fragment
  const int hi   = lane >> 4;    // lane-half selects K pair / M half

  // --- Stage x[n,:,t-tile,:] into LDS transposed, via async LDS loads ---
  // idx = c*400 + col (global-coalesced runs of 400); LDS dest Xs2[col][c].
  {
    const float* gbase = x + (size_t)n * C_ * T_ * V_ + t0 * V_;
    for (int idx = tid; idx < C_ * 400; idx += 256) {
      const int c = idx / 400, col = idx % 400;
      const float* gp = gbase + c * (T_ * V_) + col;
      const unsigned lo = lds_off(Xs2 + col * 68 + c);
      asm volatile("global_load_async_to_lds_b32 %0, %1, off"
                   :: "v"(lo), "v"((unsigned long long)(uintptr_t)gp)
                   : "memory");
    }
  }
  // Zero x2s + gs2 (+guard) once; pad lanes (v=25..27) stay zero forever.
  for (int idx = tid; idx < 16 * TT * 28 + 16 * V_ * 28 + 256; idx += 256)
    x2s[idx] = 0.f;
  asm volatile("s_wait_asynccnt 0x0" ::: "memory");
  __syncthreads();

  for (int oc = 0; oc < 4; ++oc) {
    const int o0 = oc * 16;
    v8f oacc[4];
    #pragma unroll
    for (int ui = 0; ui < 4; ++ui) oacc[ui] = (v8f){0.f,0.f,0.f,0.f,0.f,0.f,0.f,0.f};

    for (int k = 0; k < K_; ++k) {
      __syncthreads();  // prior GEMM2 reads done before we overwrite LDS
      // Load W3 chunk (padded stride) and g = graphs + A as [o][u][v28]
      for (int idx = tid; idx < 16 * C_; idx += 256)
        w3s[(idx >> 6) * 68 + (idx & 63)] = W3[(size_t)k * O_ * C_ + o0 * C_ + idx];
      const float* gbase = graphs + (((size_t)k * N_ + n) * O_ + o0) * (V_ * V_);
      const float* abase = Ag + k * (V_ * V_);
      for (int idx = tid; idx < 16 * V_ * V_; idx += 256) {
        const int o = idx / (V_ * V_), rem = idx % (V_ * V_);
        const int u = rem / V_, v = rem - u * V_;
        gs2[o * (V_ * 28) + u * 28 + v] = gbase[(size_t)o * (V_ * V_) + rem] + abase[rem];
      }
      __syncthreads();

      // --- GEMM1: x2s[o][t][v] = W3[k,o0:o0+16,:] @ x-tile  (+ b3) ---
      // 25 column tiles of 16 tv each, split across 8 waves (wave-uniform).
      for (int tile = wv; tile < 25; tile += 8) {
        const int col0 = tile * 16;
        v8f acc = (v8f){0.f,0.f,0.f,0.f,0.f,0.f,0.f,0.f};
        #pragma unroll
        for (int c0 = 0; c0 < C_; c0 += 4) {
          v2f a, b;
          // A (16x4): VGPR0 K = c0(+2 hi-half), VGPR1 K = c0+1(+2) -> adjacent
          const int ca = c0 + (hi ? 2 : 0);
          a.x = w3s[lh * 68 + ca];
          a.y = w3s[lh * 68 + ca + 1];
          // B (4x16): rows K=c, cols N=tv ; Xs2 transposed -> adjacent pair
          b.x = Xs2[(col0 + lh) * 68 + ca];
          b.y = Xs2[(col0 + lh) * 68 + ca + 1];
          acc = __builtin_amdgcn_wmma_f32_16x16x4_f32(
              false, a, false, b, (short)0, acc, false, false);
        }
        const int col = col0 + lh;
        const int t = col / 25, v = col - t * 25;
        #pragma unroll
        for (int e = 0; e < 8; ++e) {
          const int orow = e + (hi ? 8 : 0);            // D: VGPR e -> M row
          x2s[(orow * TT + t) * 28 + v] = acc[e] + b3[k * O_ + o0 + orow];
        }
      }
      __syncthreads();

      // --- GEMM2: oacc[t][u] += x2s[o] (16x28) @ g[o] (28x25 via [u][v]) ---
      // 32 units = 16 o * 2 u-tiles; 4 per wave; K over v padded to 28.
      #pragma unroll
      for (int ui = 0; ui < 4; ++ui) {
        const int unit = wv * 4 + ui;
        const int o = unit >> 1;
        const int u0 = (unit & 1) * 16;
        v8f acc2 = oacc[ui];
        #pragma unroll
        for (int v0 = 0; v0 < 28; v0 += 4) {
          v2f a, b;
          const int vr = v0 + (hi ? 2 : 0);
          a.x = x2s[(o * TT + lh) * 28 + vr];
          a.y = x2s[(o * TT + lh) * 28 + vr + 1];
          // B[v][u] = g[o][u][v]: lane=u, value pair over v -> adjacent pair
          b.x = gs2[o * (V_ * 28) + (u0 + lh) * 28 + vr];
          b.y = gs2[o * (V_ * 28) + (u0 + lh) * 28 + vr + 1];
          acc2 = __builtin_amdgcn_wmma_f32_16x16x4_f32(
              false, a, false, b, (short)0, acc2, false, false);
        }
        oacc[ui] = acc2;
      }
    }

    // Store out[n, o0+o, t0+t, u] from accumulators (u-tile 1 clipped at 25)
    #pragma unroll
    for (int ui = 0; ui < 4; ++ui) {
      const int unit = wv * 4 + ui;
      const int o = o0 + (unit >> 1);
      const int u = (unit & 1) * 16 + lh;
      if (u < V_) {
        #pragma unroll
        for (int e = 0; e < 8; ++e) {
          const int t = e + (hi ? 8 : 0);
          out[(((size_t)n * O_ + o) * T_ + t0 + t) * V_ + u] = oacc[ui][e];
        }
      }
    }
  }
}

// ---------------------------------------------------------------------------
extern "C" void kernel_launch(void* const* d_in, const int* in_sizes, int n_in,
                              void* d_out, int out_size, void* d_ws, size_t ws_size,
                              hipStream_t stream) {
  const float* x   = (const float*)d_in[0];
  const float* Ag  = (const float*)d_in[1];
  const float* W11 = (const float*)d_in[2];
  const float* b11 = (const float*)d_in[3];
  const float* W12 = (const float*)d_in[4];
  const float* b12 = (const float*)d_in[5];
  const float* W2  = (const float*)d_in[6];
  const float* b2  = (const float*)d_in[7];
  const float* W3  = (const float*)d_in[8];
  const float* b3  = (const float*)d_in[9];
  const float* W4  = (const float*)d_in[10];
  const float* b4  = (const float*)d_in[11];

  float* out    = (float*)d_out;                       // (N,O,T,V)
  float* graphs = out + (size_t)N_ * O_ * T_ * V_;     // (K,N,O,25,25)
  float* pooled = (float*)d_ws;                        // (N,C,25) scratch

  pool_kernel<<<N_ * C_, 256, 0, stream>>>(x, pooled);
  graphs_kernel<<<dim3(N_, K_), 256, 0, stream>>>(
      pooled, Ag, W11, b11, W12, b12, W2, b2, W4, b4, graphs);

  const size_t smemC = (size_t)(400 * 68 + 16 * 16 * 28 + 16 * V_ * 28 + 256 + 16 * 68)
                       * sizeof(float);
  hipFuncSetAttribute((const void*)fused_gemm_kernel,
                      hipFuncAttributeMaxDynamicSharedMemorySize, (int)smemC);
  fused_gemm_kernel<<<dim3(T_ / 16, N_), 256, smemC, stream>>>(
      x, graphs, Ag, W3, b3, out);
}